// MAGNN_18949395710220
// MI455X (gfx1250) — compile-verified
//
#include <hip/hip_runtime.h>
#include <hip/hip_bf16.h>
#include <math.h>

// MAGNN forward for MI455X (gfx1250, wave32).
// Memory-bound (~360MB traffic vs ~30 GFLOP) -> keep f32, use
// V_WMMA_F32_16X16X4_F32 for the three 128-K GEMMs and the Att@H GEMM.

typedef __attribute__((ext_vector_type(2))) float v2f;
typedef __attribute__((ext_vector_type(8))) float v8f;

#define B_DIM 4096
#define L_DIM 50
#define T_DIM 100
#define D1 128
#define LP 64            // L padded to multiple of 16
#define NEG_INF (-9000000000000000.0f)

// H(64x128, LDS) = X(64x128, LDS) @ Wg(128x128, global row-major)
// optionally: H = tanh(H + bias[col])
template <bool TANHB>
__device__ __forceinline__ void gemm_xw(const float* __restrict__ Wg,
                                        const float* __restrict__ bias,
                                        const float* Xl, float* Hl,
                                        int lane, int wave) {
  const int lrow = lane & 15;
  const int lhi  = lane >> 4;
  const int nt   = wave;                 // 8 waves -> 8 N tiles
  const int bcol = nt * 16 + lrow;
#pragma unroll
  for (int mt = 0; mt < 4; ++mt) {       // 4 M tiles (64 rows)
    v8f acc = {};
    const int arow = mt * 16 + lrow;
    for (int k0 = 0; k0 < D1; k0 += 4) {
      const int kk = k0 + lhi * 2;
      v2f a;  a.x = Xl[arow * D1 + kk];        a.y = Xl[arow * D1 + kk + 1];
      v2f bb; bb.x = Wg[kk * D1 + bcol];       bb.y = Wg[(kk + 1) * D1 + bcol];
      acc = __builtin_amdgcn_wmma_f32_16x16x4_f32(false, a, false, bb,
                                                  (short)0, acc, false, false);
    }
    const int drow = mt * 16 + lhi * 8;
#pragma unroll
    for (int r = 0; r < 8; ++r) {
      float v = acc[r];
      if constexpr (TANHB) v = tanhf(v + bias[bcol]);
      Hl[(drow + r) * D1 + bcol] = v;
    }
  }
}

// X(64x128) = elu( Att(64x64, LDS) @ H(64x128, LDS) )
__device__ __forceinline__ void gemm_atth(const float* Att, const float* Hl,
                                          float* Xl, int lane, int wave) {
  const int lrow = lane & 15;
  const int lhi  = lane >> 4;
  const int nt   = wave;
  const int bcol = nt * 16 + lrow;
#pragma unroll
  for (int mt = 0; mt < 4; ++mt) {
    v8f acc = {};
    const int arow = mt * 16 + lrow;
    for (int k0 = 0; k0 < LP; k0 += 4) {
      const int kk = k0 + lhi * 2;
      v2f a;  a.x = Att[arow * LP + kk];   a.y = Att[arow * LP + kk + 1];
      v2f bb; bb.x = Hl[kk * D1 + bcol];   bb.y = Hl[(kk + 1) * D1 + bcol];
      acc = __builtin_amdgcn_wmma_f32_16x16x4_f32(false, a, false, bb,
                                                  (short)0, acc, false, false);
    }
    const int drow = mt * 16 + lhi * 8;
#pragma unroll
    for (int r = 0; r < 8; ++r) {
      float v = acc[r];
      v = v > 0.f ? v : (expf(v) - 1.f);   // elu
      Xl[(drow + r) * D1 + bcol] = v;
    }
  }
}

__global__ __launch_bounds__(256) void MAGNN_18949395710220_kernel(
    const int* __restrict__ item_seq, const int* __restrict__ user_ids,
    const int* __restrict__ items_to_predict, const int* __restrict__ A,
    const float* __restrict__ item_emb, const float* __restrict__ user_emb,
    const float* __restrict__ W2_table, const float* __restrict__ b2_table,
    const float* __restrict__ W_att, const float* __restrict__ a_att,
    const float* __restrict__ W_out, const float* __restrict__ a_out,
    const float* __restrict__ att1_W, const float* __restrict__ att1_b,
    const float* __restrict__ att2_W, const float* __restrict__ att2_b,
    const float* __restrict__ user_com, float* __restrict__ out) {
  __shared__ float Xl[LP * D1];     // 32 KB
  __shared__ float Hl[LP * D1];     // 32 KB
  __shared__ float Att[LP * LP];    // 16 KB
  __shared__ float f1[LP], f2[LP];
  __shared__ float attnL[LP * 4];
  __shared__ float svec[D1];        // sum_l item_embs
  __shared__ float aevec[D1];       // attention_embs
  __shared__ float fsvec[D1];       // fusion + svec

  const int b    = blockIdx.x;
  const int tid  = threadIdx.x;
  const int lane = tid & 31;
  const int wave = tid >> 5;

  // ---- gather item embeddings, pad rows [50,64) with zero ----
  const int* seq = item_seq + b * L_DIM;
  for (int idx = tid; idx < LP * D1; idx += 256) {
    const int row = idx >> 7, col = idx & (D1 - 1);
    float v = 0.f;
    if (row < L_DIM) v = item_emb[(long long)seq[row] * D1 + col];
    Xl[idx] = v;
  }
  __syncthreads();
  if (tid < D1) {                    // svec[d] = sum_l item_embs[l][d]
    float acc = 0.f;
    for (int l = 0; l < L_DIM; ++l) acc += Xl[l * D1 + tid];
    svec[tid] = acc;
  }
  __syncthreads();

  const int* adj = A + b * (L_DIM * L_DIM);

  // ---- two GAT layers ----
  for (int layer = 0; layer < 2; ++layer) {
    const float* Wg = (layer == 0) ? W_att : W_out;
    const float* ag = (layer == 0) ? a_att : a_out;

    gemm_xw<false>(Wg, nullptr, Xl, Hl, lane, wave);   // H = X @ W
    __syncthreads();

    if (tid < L_DIM) {                                  // f1 = H @ a1
      float acc = 0.f;
      for (int d = 0; d < D1; ++d) acc += Hl[tid * D1 + d] * ag[d];
      f1[tid] = acc;
    } else if (tid >= 64 && tid < 64 + L_DIM) {         // f2 = H @ a2
      const int j = tid - 64;
      float acc = 0.f;
      for (int d = 0; d < D1; ++d) acc += Hl[j * D1 + d] * ag[D1 + d];
      f2[j] = acc;
    }
    __syncthreads();

    if (tid < LP) {                                     // masked softmax rows
      const int i = tid;
      if (i < L_DIM) {
        float rmax = -3.4e38f;
        for (int j = 0; j < L_DIM; ++j) {
          const float e = tanhf(f1[i] + f2[j]);
          const float m = (adj[i * L_DIM + j] > 0) ? e : NEG_INF;
          Att[i * LP + j] = m;
          rmax = fmaxf(rmax, m);
        }
        float sum = 0.f;
        for (int j = 0; j < L_DIM; ++j) {
          const float p = expf(Att[i * LP + j] - rmax);  // all-masked -> uniform, matches jax
          Att[i * LP + j] = p;
          sum += p;
        }
        const float inv = 1.f / sum;
        for (int j = 0; j < L_DIM; ++j) Att[i * LP + j] *= inv;
        for (int j = L_DIM; j < LP; ++j) Att[i * LP + j] = 0.f;
      } else {
        for (int j = 0; j < LP; ++j) Att[i * LP + j] = 0.f;
      }
    }
    __syncthreads();

    gemm_atth(Att, Hl, Xl, lane, wave);                 // X = elu(Att @ H)
    __syncthreads();
  }

  // ---- attention pooling head ----
  gemm_xw<true>(att1_W, att1_b, Xl, Hl, lane, wave);    // M1 = tanh(X@att1_W + b)
  __syncthreads();

  if (tid < L_DIM) {                                    // m2 + softmax over H=4
    float m2[4];
#pragma unroll
    for (int h = 0; h < 4; ++h) {
      float acc = att2_b[h];
      for (int d = 0; d < D1; ++d) acc += Hl[tid * D1 + d] * att2_W[d * 4 + h];
      m2[h] = acc;
    }
    const float mx = fmaxf(fmaxf(m2[0], m2[1]), fmaxf(m2[2], m2[3]));
    float sum = 0.f;
#pragma unroll
    for (int h = 0; h < 4; ++h) { m2[h] = expf(m2[h] - mx); sum += m2[h]; }
    const float inv = 1.f / sum;
#pragma unroll
    for (int h = 0; h < 4; ++h) attnL[tid * 4 + h] = m2[h] * inv;
  }
  __syncthreads();

  if (tid < D1) {                                       // attention_embs
    const int d = tid;
    float acc = 0.f;
#pragma unroll
    for (int h = 0; h < 4; ++h) {
      float z = 0.f;
      for (int l = 0; l < L_DIM; ++l) z += Xl[l * D1 + d] * attnL[l * 4 + h];
      acc += tanhf(z);
    }
    aevec[d] = acc * 0.25f;
  }
  __syncthreads();

  const float* ue = user_emb + (long long)user_ids[b] * D1;
  if (tid < D1) {                                       // fusion + svec
    const int g = tid;
    float acc = 0.f;
    for (int k = 0; k < D1; ++k) acc += aevec[k] * user_com[k * D1 + g];
    for (int k = 0; k < D1; ++k) acc += ue[k] * user_com[(D1 + k) * D1 + g];
    fsvec[g] = acc + svec[g];
  }
  __syncthreads();

  // ---- scoring: one wave32 per target item ----
  const int* itp = items_to_predict + b * T_DIM;
  for (int t = wave; t < T_DIM; t += 8) {
    const int item = itp[t];
    const float* w2r = W2_table + (long long)item * D1;
    float partial = 0.f;
    for (int d = lane; d < D1; d += 32) partial += w2r[d] * fsvec[d];
#pragma unroll
    for (int off = 16; off > 0; off >>= 1)
      partial += __shfl_down(partial, off, 32);
    if (lane == 0) out[b * T_DIM + t] = partial + b2_table[item];
  }
}

extern "C" void kernel_launch(void* const* d_in, const int* in_sizes, int n_in,
                              void* d_out, int out_size, void* d_ws, size_t ws_size,
                              hipStream_t stream) {
  (void)in_sizes; (void)n_in; (void)out_size; (void)d_ws; (void)ws_size;
  const int*   item_seq    = (const int*)  d_in[0];
  const int*   user_ids    = (const int*)  d_in[1];
  const int*   items_pred  = (const int*)  d_in[2];
  const int*   A           = (const int*)  d_in[3];
  const float* item_emb    = (const float*)d_in[4];
  const float* user_emb    = (const float*)d_in[5];
  const float* W2_table    = (const float*)d_in[6];
  const float* b2_table    = (const float*)d_in[7];
  const float* W_att       = (const float*)d_in[8];
  const float* a_att       = (const float*)d_in[9];
  const float* W_out       = (const float*)d_in[10];
  const float* a_out       = (const float*)d_in[11];
  const float* att1_W      = (const float*)d_in[12];
  const float* att1_b      = (const float*)d_in[13];
  const float* att2_W      = (const float*)d_in[14];
  const float* att2_b      = (const float*)d_in[15];
  const float* user_com    = (const float*)d_in[16];
  float* out = (float*)d_out;

  MAGNN_18949395710220_kernel<<<B_DIM, 256, 0, stream>>>(
      item_seq, user_ids, items_pred, A, item_emb, user_emb, W2_table, b2_table,
      W_att, a_att, W_out, a_out, att1_W, att1_b, att2_W, att2_b, user_com, out);
}